// VARS_SD_18098992185937
// MI455X (gfx1250) — compile-verified
//
#include <hip/hip_runtime.h>
#include <hip/hip_bf16.h>
#include <math.h>

// ---------------------------------------------------------------------------
// Problem constants (from reference)
// ---------------------------------------------------------------------------
constexpr int  CB   = 32;    // batch
constexpr int  CN   = 576;   // tokens
constexpr int  CDIM = 1024;  // model dim
constexpr int  CH   = 16;    // heads
constexpr int  CHD  = 64;    // head dim
constexpr int  CM   = 128;   // rand feat dim
constexpr int  CP   = 256;   // 2*M
constexpr int  NUM_STEP = 10;
constexpr long QH_SZ = (long)CB * CH * CN * CHD;              // 18,874,368
constexpr float SQRT_SCALE = 0.35355339059327379f;            // (HD^-0.5)^0.5
constexpr float RSQRT_M    = 0.08838834764831845f;            // 1/sqrt(128)
constexpr float LAMC       = 0.1f;

typedef __attribute__((ext_vector_type(2))) float v2f;
typedef __attribute__((ext_vector_type(8))) float v8f;

// Exact-fp32 WMMA: D(16x16) = A(16x4) * B(4x16) + C
__device__ __forceinline__ v8f wmma4(v2f a, v2f b, v8f c) {
  return __builtin_amdgcn_wmma_f32_16x16x4_f32(
      /*neg_a=*/false, a, /*neg_b=*/false, b,
      /*c_mod=*/(short)0, c, /*reuse_a=*/false, /*reuse_b=*/false);
}

// sign(z) * gelu_exact(|z| - t)
__device__ __forceinline__ float soft_thresh(float z, float t) {
  float a = fabsf(z) - t;
  float g = 0.5f * a * (1.0f + erff(a * 0.70710678118654752f));
  return z > 0.f ? g : (z < 0.f ? -g : 0.f);
}

// ---------------------------------------------------------------------------
// Generic batched fp32 WMMA GEMM.
//   C[b] = op(A[b]) * op(B[b])   (op = optional transpose via TA/TB)
// 256 threads = 8 waves; 128x128 workgroup tile; K staged via LDS in chunks
// of 16 (4 wmma k-steps). Wave w: rows 32*(w&3), cols 64*(w>>2); 2x4 tiles.
// EPI: 0 = plain (+optional bias), 1 = qkv head scatter (+sqrt(SCALE) on q),
//      2 = Performer phi: exp(val - aux[row]) / sqrt(M)
// Batch offsets: off = (batch/cnt)*outer + (batch%cnt)*inner
// ---------------------------------------------------------------------------
template<bool TA, bool TB, int EPI>
__global__ void __launch_bounds__(256) gemm_wmma(
    const float* __restrict__ A, const float* __restrict__ B,
    float* __restrict__ C, const float* __restrict__ bias,
    const float* __restrict__ aux,
    int Mr, int Nc, int K,
    int lda, int ldb, int ldc,
    long aOuter, long aInner, int aCnt,
    long bOuter, long bInner, int bCnt,
    long cOuter, long cInner, int cCnt,
    long auxStride)
{
  __shared__ float As[16][132];
  __shared__ float Bs[16][132];

  const int batch = blockIdx.z;
  const long aOff = (long)(batch / aCnt) * aOuter + (long)(batch % aCnt) * aInner;
  const long bOff = (long)(batch / bCnt) * bOuter + (long)(batch % bCnt) * bInner;
  const long cOff = (long)(batch / cCnt) * cOuter + (long)(batch % cCnt) * cInner;

  const int tileM = blockIdx.x * 128;
  const int tileN = blockIdx.y * 128;
  const int tid   = threadIdx.x;
  const int lane  = tid & 31;
  const int wave  = tid >> 5;
  const int lhalf = lane >> 4;      // 0: lanes 0-15, 1: lanes 16-31
  const int lm    = lane & 15;
  const int wm    = wave & 3;       // M quadrant (rows 32*wm)
  const int wn    = wave >> 2;      // N half     (cols 64*wn)

  v8f acc[2][4] = {};

  for (int kc = 0; kc < K; kc += 16) {
    // --- stage op(A) tile as As[k][m], coalesced along the stored-fast dim
    for (int i = tid; i < 2048; i += 256) {
      int m, kk;
      if (TA) { m = i & 127; kk = i >> 7; } else { kk = i & 15; m = i >> 4; }
      int gm = tileM + m, gk = kc + kk;
      float v = 0.f;
      if (gm < Mr && gk < K)
        v = TA ? A[aOff + (long)gk * lda + gm] : A[aOff + (long)gm * lda + gk];
      As[kk][m] = v;
    }
    // --- stage op(B) tile as Bs[k][n]
    for (int i = tid; i < 2048; i += 256) {
      int n, kk;
      if (TB) { kk = i & 15; n = i >> 4; } else { n = i & 127; kk = i >> 7; }
      int gn = tileN + n, gk = kc + kk;
      float v = 0.f;
      if (gn < Nc && gk < K)
        v = TB ? B[bOff + (long)gn * ldb + gk] : B[bOff + (long)gk * ldb + gn];
      Bs[kk][n] = v;
    }
    __syncthreads();

    #pragma unroll
    for (int ks = 0; ks < 4; ++ks) {
      const int k0 = 4 * ks + 2 * lhalf;   // ISA layout: lanes>=16 hold k+2,k+3
      v2f af[2], bf[4];
      #pragma unroll
      for (int mt = 0; mt < 2; ++mt) {
        int mm = 32 * wm + 16 * mt + lm;
        af[mt].x = As[k0][mm];
        af[mt].y = As[k0 + 1][mm];
      }
      #pragma unroll
      for (int nt = 0; nt < 4; ++nt) {
        int nn = 64 * wn + 16 * nt + lm;
        bf[nt].x = Bs[k0][nn];
        bf[nt].y = Bs[k0 + 1][nn];
      }
      #pragma unroll
      for (int mt = 0; mt < 2; ++mt)
        #pragma unroll
        for (int nt = 0; nt < 4; ++nt)
          acc[mt][nt] = wmma4(af[mt], bf[nt], acc[mt][nt]);
    }
    __syncthreads();
  }

  // --- epilogue (C/D layout: row = v + 8*lhalf, col = lm within tile)
  #pragma unroll
  for (int mt = 0; mt < 2; ++mt)
  #pragma unroll
  for (int nt = 0; nt < 4; ++nt)
  #pragma unroll
  for (int v = 0; v < 8; ++v) {
    int row = tileM + 32 * wm + 16 * mt + v + 8 * lhalf;
    int col = tileN + 64 * wn + 16 * nt + lm;
    if (row >= Mr || col >= Nc) continue;
    float val = acc[mt][nt][v];
    if (EPI == 0) {
      if (bias) val += bias[col];
      C[cOff + (long)row * ldc + col] = val;
    } else if (EPI == 1) {
      // qkv: row = b*N+n, col = h*HD+d; batch 0 -> q (scaled), 1 -> v
      int bb = row / CN, n = row % CN;
      int h  = col / CHD, d = col % CHD;
      float s = (batch == 0) ? SQRT_SCALE : 1.0f;
      C[(long)batch * QH_SZ + (((long)bb * CH + h) * CN + n) * CHD + d] = val * s;
    } else { // EPI == 2: Performer positive feature map
      float e = expf(val - aux[auxStride * batch + row]) * RSQRT_M;
      C[cOff + (long)row * ldc + col] = e;
    }
  }
}

// ---------------------------------------------------------------------------
// rnorm[bh,n] = 0.5 * sum_d q[bh,n,d]^2   (q already scaled by sqrt(SCALE))
// ---------------------------------------------------------------------------
__global__ void rnorm_kernel(const float* __restrict__ qh,
                             float* __restrict__ rn, int tot) {
  int idx = blockIdx.x * blockDim.x + threadIdx.x;
  if (idx >= tot) return;
  const float* row = qh + (long)idx * CHD;
  float s = 0.f;
  #pragma unroll 8
  for (int d = 0; d < CHD; ++d) { float v = row[d]; s += v * v; }
  rn[idx] = 0.5f * s;
}

// qcat[bh,n,128+m] = kernel[h,n,m] (broadcast over b)
__global__ void fillk_kernel(const float* __restrict__ kern,
                             float* __restrict__ qcat) {
  long idx = (long)blockIdx.x * blockDim.x + threadIdx.x;
  const long tot = (long)CB * CH * CN * CM;
  if (idx >= tot) return;
  int m  = (int)(idx % CM);
  long t = idx / CM;
  int n  = (int)(t % CN);
  int bh = (int)(t / CN);
  int h  = bh % CH;
  qcat[((long)bh * CN + n) * CP + CM + m] = kern[((long)h * CN + n) * CM + m];
}

// L2-normalize qcat over the token dim (per (bh, column p))
__global__ void normalize_kernel(float* __restrict__ qcat) {
  int bh = blockIdx.x;
  float* qb = qcat + (long)bh * CN * CP;
  int p = threadIdx.x;           // 256 threads = 256 columns, coalesced
  float s = 0.f;
  for (int n = 0; n < CN; ++n) { float v = qb[(long)n * CP + p]; s += v * v; }
  float inv = 1.f / fmaxf(sqrtf(s), 1e-12f);
  for (int n = 0; n < CN; ++n) qb[(long)n * CP + p] *= inv;
}

// L[bh] = max_p sum_q |kk[p][q]| + 1
__global__ void lmax_kernel(const float* __restrict__ kk,
                            float* __restrict__ Lbuf) {
  __shared__ float red[CP];
  int bh = blockIdx.x;
  const float* kb = kk + (long)bh * CP * CP;
  int p = threadIdx.x;
  float s = 0.f;
  for (int q = 0; q < CP; ++q) s += fabsf(kb[(long)p * CP + q]);
  red[p] = s;
  __syncthreads();
  for (int st = CP / 2; st > 0; st >>= 1) {
    if (p < st) red[p] = fmaxf(red[p], red[p + st]);
    __syncthreads();
  }
  if (p == 0) Lbuf[bh] = red[0] + 1.0f;
}

// ---------------------------------------------------------------------------
// ISTA solver: one workgroup (16 waves) per (b,h).
//   z in LDS (64KB) doubles as the WMMA B operand; inp & z register copies
//   held in WMMA C-layout; kk_learned streamed from L2 as A fragments.
// ---------------------------------------------------------------------------
__global__ void __launch_bounds__(512) ista_kernel(
    const float* __restrict__ kkL, const float* __restrict__ inp,
    float* __restrict__ zout, const float* __restrict__ Lbuf,
    const float* __restrict__ lamPtr, const float* __restrict__ Lsc)
{
  __shared__ float zl[CP][CHD];             // 256*64*4 = 64 KB
  const int bh    = blockIdx.x;
  const int tid   = threadIdx.x;
  const int lane  = tid & 31;
  const int wave  = tid >> 5;               // 16 waves -> 16 row-tiles
  const int lhalf = lane >> 4;
  const int lm    = lane & 15;
  const int m0    = wave * 16;

  const float* kb = kkL + (long)bh * CP * CP;
  const float* ib = inp + (long)bh * CP * CHD;
  const float lamL = lamPtr[0] * LAMC;
  const float effL = Lbuf[bh] / Lsc[0];
  const float invL = 1.0f / effL;
  const float thr  = lamL * invL;

  float cin[4][8], z[4][8];
  #pragma unroll
  for (int nt = 0; nt < 4; ++nt)
    #pragma unroll
    for (int v = 0; v < 8; ++v) {
      int r = m0 + v + 8 * lhalf;
      int c = 16 * nt + lm;
      float t = ib[(long)r * CHD + c];
      cin[nt][v] = t;
      float zz = soft_thresh(t, lamL);      // z0 = soft(inp, lam)
      z[nt][v] = zz;
      zl[r][c] = zz;
    }
  __syncthreads();

  for (int it = 0; it < NUM_STEP; ++it) {
    v8f acc[4] = {};
    for (int k = 0; k < CP; k += 4) {
      const int ka = k + 2 * lhalf;
      v2f a;
      a.x = kb[(long)(m0 + lm) * CP + ka];
      a.y = kb[(long)(m0 + lm) * CP + ka + 1];
      #pragma unroll
      for (int nt = 0; nt < 4; ++nt) {
        v2f bfr;
        bfr.x = zl[ka][16 * nt + lm];
        bfr.y = zl[ka + 1][16 * nt + lm];
        acc[nt] = wmma4(a, bfr, acc[nt]);
      }
    }
    __syncthreads();                         // all reads of zl done
    #pragma unroll
    for (int nt = 0; nt < 4; ++nt)
      #pragma unroll
      for (int v = 0; v < 8; ++v) {
        int r = m0 + v + 8 * lhalf;
        int c = 16 * nt + lm;
        float w  = acc[nt][v];
        float zn = z[nt][v] - (w - cin[nt][v]) * invL;
        zn = soft_thresh(zn, thr);
        z[nt][v] = zn;
        zl[r][c] = zn;
      }
    __syncthreads();                         // zl updated before next read
  }

  float* zb = zout + (long)bh * CP * CHD;
  #pragma unroll
  for (int nt = 0; nt < 4; ++nt)
    #pragma unroll
    for (int v = 0; v < 8; ++v) {
      int r = m0 + v + 8 * lhalf;
      int c = 16 * nt + lm;
      zb[(long)r * CHD + c] = z[nt][v];
    }
}

// ---------------------------------------------------------------------------
// Host orchestration
// ---------------------------------------------------------------------------
extern "C" void kernel_launch(void* const* d_in, const int* in_sizes, int n_in,
                              void* d_out, int out_size, void* d_ws, size_t ws_size,
                              hipStream_t stream) {
  (void)in_sizes; (void)n_in; (void)out_size; (void)ws_size;

  const float* x     = (const float*)d_in[0];
  const float* qkvw  = (const float*)d_in[1];
  const float* projw = (const float*)d_in[2];
  const float* projb = (const float*)d_in[3];
  const float* randm = (const float*)d_in[4];
  const float* kern  = (const float*)d_in[5];
  const float* lk    = (const float*)d_in[6];
  const float* llam  = (const float*)d_in[7];
  const float* lL    = (const float*)d_in[8];
  float* out = (float*)d_out;

  float* ws = (float*)d_ws;
  long off = 0;
  float* qh    = ws + off; off += QH_SZ;                         // q heads; reused as y
  float* vh    = ws + off; off += QH_SZ;                         // v heads
  float* qcat  = ws + off; off += (long)CB * CH * CN * CP;       // [bh,N,256]
  float* rnorm = ws + off; off += (long)CB * CH * CN;            // 0.5*|q|^2
  float* kk    = ws + off; off += (long)CB * CH * CP * CP;       // kk -> kk_learned
  float* kk2   = ws + off; off += (long)CB * CH * CP * CP;       // K@kk temp
  float* inp   = ws + off; off += (long)CB * CH * CP * CHD;
  float* zbuf  = ws + off; off += (long)CB * CH * CP * CHD;
  float* Lbuf  = ws + off; off += CB * CH;

  const int BN = CB * CN;   // 18432
  const int BH = CB * CH;   // 512

  // 1) qkv GEMM: C = x @ qkv_w^T, batched over {q,v}, fused head scatter+scale
  {
    dim3 g((BN + 127) / 128, (CDIM + 127) / 128, 2);
    gemm_wmma<false, true, 1><<<g, 256, 0, stream>>>(
        x, qkvw, qh, nullptr, nullptr,
        BN, CDIM, CDIM, CDIM, CDIM, 0,
        0, 0, 1, (long)CDIM * CDIM, 0, 1, 0, 0, 1, 0);
  }
  // 2) per-row 0.5*|q|^2
  { int tot = BH * CN;
    rnorm_kernel<<<(tot + 255) / 256, 256, 0, stream>>>(qh, rnorm, tot); }
  // 3) phi = exp(q@R - rnorm)/sqrt(M) -> qcat[:, :128]
  {
    dim3 g((CN + 127) / 128, 1, BH);
    gemm_wmma<false, false, 2><<<g, 256, 0, stream>>>(
        qh, randm, qcat, nullptr, rnorm,
        CN, CM, CHD, CHD, CM, CP,
        (long)CN * CHD, 0, 1,
        0, (long)CHD * CM, CH,
        (long)CN * CP, 0, 1, CN);
  }
  // 4) qcat[:, 128:] = learned kernel features
  { long tot = (long)BH * CN * CM;
    fillk_kernel<<<(int)((tot + 255) / 256), 256, 0, stream>>>(kern, qcat); }
  // 5) L2-normalize over tokens
  normalize_kernel<<<BH, CP, 0, stream>>>(qcat);
  // 6) kk = Qc^T @ Qc
  {
    dim3 g(2, 2, BH);
    gemm_wmma<true, false, 0><<<g, 256, 0, stream>>>(
        qcat, qcat, kk, nullptr, nullptr,
        CP, CP, CN, CP, CP, CP,
        (long)CN * CP, 0, 1, (long)CN * CP, 0, 1, (long)CP * CP, 0, 1, 0);
  }
  // 7) inp = Qc^T @ V
  {
    dim3 g(2, 1, BH);
    gemm_wmma<true, false, 0><<<g, 256, 0, stream>>>(
        qcat, vh, inp, nullptr, nullptr,
        CP, CHD, CN, CP, CHD, CHD,
        (long)CN * CP, 0, 1, (long)CN * CHD, 0, 1, (long)CP * CHD, 0, 1, 0);
  }
  // 8) Lipschitz bound L = max row-sum(|kk|) + 1  (before learned_k transform)
  lmax_kernel<<<BH, CP, 0, stream>>>(kk, Lbuf);
  // 9) kk2 = learned_k @ kk
  {
    dim3 g(2, 2, BH);
    gemm_wmma<false, false, 0><<<g, 256, 0, stream>>>(
        lk, kk, kk2, nullptr, nullptr,
        CP, CP, CP, CP, CP, CP,
        0, 0, 1, (long)CP * CP, 0, 1, (long)CP * CP, 0, 1, 0);
  }
  // 10) kk_learned = kk2 @ learned_k  (overwrite kk)
  {
    dim3 g(2, 2, BH);
    gemm_wmma<false, false, 0><<<g, 256, 0, stream>>>(
        kk2, lk, kk, nullptr, nullptr,
        CP, CP, CP, CP, CP, CP,
        (long)CP * CP, 0, 1, 0, 0, 1, (long)CP * CP, 0, 1, 0);
  }
  // 11) ISTA unrolled solver
  ista_kernel<<<BH, 512, 0, stream>>>(kk, inp, zbuf, Lbuf, llam, lL);
  // 12) out heads: y[b,n,h*HD+d] = Qc @ z   (y reuses qh buffer)
  {
    dim3 g((CN + 127) / 128, 1, BH);
    gemm_wmma<false, false, 0><<<g, 256, 0, stream>>>(
        qcat, zbuf, qh, nullptr, nullptr,
        CN, CHD, CP, CP, CHD, CDIM,
        (long)CN * CP, 0, 1, (long)CP * CHD, 0, 1,
        (long)CN * CDIM, (long)CHD, CH, 0);
  }
  // 13) final projection: out = y @ proj_w^T + proj_b
  {
    dim3 g((BN + 127) / 128, (CDIM + 127) / 128, 1);
    gemm_wmma<false, true, 0><<<g, 256, 0, stream>>>(
        qh, projw, out, projb, nullptr,
        BN, CDIM, CDIM, CDIM, CDIM, CDIM,
        0, 0, 1, 0, 0, 1, 0, 0, 1, 0);
  }
}